// CollisionAccuracy_41721312313866
// MI455X (gfx1250) — compile-verified
//
#include <hip/hip_runtime.h>

typedef __attribute__((ext_vector_type(2))) float v2f;
typedef __attribute__((ext_vector_type(8))) float v8f;

#define B_    4
#define NQ_   8192
#define NA_   6890
#define NCH   ((NA_ + 15) / 16)     // 431 chunks of 16 anchors
#define NAP   (NCH * 16)            // 6896 padded anchors
#define WPB   8                     // waves per block
#define TPB   (NQ_ / 16)            // 512 query tiles per batch
#define MAXD  0.5f

__device__ __forceinline__ int bperm_i(int srcLane, int v) {
  return __builtin_amdgcn_ds_bpermute(srcLane << 2, v);
}
__device__ __forceinline__ float bperm_f(int srcLane, float v) {
  return __int_as_float(__builtin_amdgcn_ds_bpermute(srcLane << 2, __float_as_int(v)));
}

__global__ void __launch_bounds__(256)
collision_count_wmma(const float* __restrict__ q,
                     const float* __restrict__ a,
                     const float* __restrict__ nrm,
                     float* __restrict__ out) {
  extern __shared__ float lds[];            // NAP * 4 floats = 110,336 bytes
  const int tid   = threadIdx.x;
  const int lane  = tid & 31;
  const int wave  = tid >> 5;
  const int bid   = blockIdx.x;
  const int b     = bid / (TPB / WPB);      // batch index (bid / 64)
  const int tile0 = (bid % (TPB / WPB)) * WPB;

  const float* qb = q + (size_t)b * NQ_ * 3;
  const float* ab = a + (size_t)b * NA_ * 3;

  // ---- Stage augmented anchor vectors w = (-2ax,-2ay,-2az, |a|^2) into LDS ----
  for (int i = tid; i < NAP; i += 256) {
    float w0 = 0.f, w1 = 0.f, w2 = 0.f, w3 = 1e30f;   // pad: huge distance
    if (i < NA_) {
      float ax = ab[i * 3 + 0], ay = ab[i * 3 + 1], az = ab[i * 3 + 2];
      w0 = -2.f * ax; w1 = -2.f * ay; w2 = -2.f * az;
      w3 = ax * ax + ay * ay + az * az;
    }
    lds[i * 4 + 0] = w0; lds[i * 4 + 1] = w1;
    lds[i * 4 + 2] = w2; lds[i * 4 + 3] = w3;
  }
  __syncthreads();

  // ---- A operand: 16 queries x (qx,qy,qz,1), loaded once per wave ----
  const int qtile = tile0 + wave;
  const int q0    = qtile * 16;
  const int row   = lane & 15;
  v2f amat;
  if (lane < 16) {
    amat.x = qb[(q0 + row) * 3 + 0];
    amat.y = qb[(q0 + row) * 3 + 1];
  } else {
    amat.x = qb[(q0 + row) * 3 + 2];
    amat.y = 1.0f;
  }

  float bestv[8];
  int   besti[8];
#pragma unroll
  for (int v = 0; v < 8; ++v) { bestv[v] = 3.4e38f; besti[v] = 0; }

  const int col  = lane & 15;
  const int half = lane >> 4;
  const v8f zero = {};

  // ---- Main scan: one WMMA per 16x16 tile of (-2 q.a + |a|^2) ----
  for (int ch = 0; ch < NCH; ++ch) {
    const int n = ch * 16 + col;
    const float* wp = &lds[n * 4 + half * 2];   // bank-conflict-free ds_load_b64
    v2f bmat; bmat.x = wp[0]; bmat.y = wp[1];
    v8f d = __builtin_amdgcn_wmma_f32_16x16x4_f32(
        false, amat, false, bmat, (short)0, zero, false, false);
#pragma unroll
    for (int v = 0; v < 8; ++v) {
      bool better = d[v] < bestv[v];
      bestv[v] = better ? d[v] : bestv[v];
      besti[v] = better ? n    : besti[v];
    }
  }

  // ---- Cross-lane argmin within each 16-lane half (tie -> lowest index) ----
#pragma unroll
  for (int m = 1; m <= 8; m <<= 1) {
    const int src = lane ^ m;
#pragma unroll
    for (int v = 0; v < 8; ++v) {
      float pv = bperm_f(src, bestv[v]);
      int   pi = bperm_i(src, besti[v]);
      bool take = (pv < bestv[v]) || (pv == bestv[v] && pi < besti[v]);
      bestv[v] = take ? pv : bestv[v];
      besti[v] = take ? pi : besti[v];
    }
  }

  // ---- Transpose: lane m (0..15) gets NN index of query row m ----
  int myIdx = 0;
#pragma unroll
  for (int v = 0; v < 8; ++v) {
    int iLo = bperm_i(0,  besti[v]);   // rows 0..7 live in lower half
    int iHi = bperm_i(16, besti[v]);   // rows 8..15 live in upper half
    if (lane == v)     myIdx = iLo;
    if (lane == v + 8) myIdx = iHi;
  }

  // ---- Epilogue: diff / norm / dot / mask / collision ----
  float contrib = 0.f;
  if (lane < 16) {
    const int qi = q0 + lane;
    float qx = qb[qi * 3 + 0], qy = qb[qi * 3 + 1], qz = qb[qi * 3 + 2];
    const float* ap = ab + (size_t)myIdx * 3;
    const float* np = nrm + ((size_t)b * NA_ + myIdx) * 3;
    float dx = qx - ap[0], dy = qy - ap[1], dz = qz - ap[2];
    float l2 = sqrtf(dx * dx + dy * dy + dz * dz);
    float dt = dx * np[0] + dy * np[1] + dz * np[2];
    dt = (l2 <= MAXD) ? dt : 0.f;
    contrib = (dt < 0.f) ? 1.f : 0.f;
  }
  // wave-wide sum, one atomic per wave
#pragma unroll
  for (int m = 1; m < 32; m <<= 1)
    contrib += bperm_f(lane ^ m, contrib);
  if (lane == 0) atomicAdd(out + b, contrib);
}

extern "C" void kernel_launch(void* const* d_in, const int* in_sizes, int n_in,
                              void* d_out, int out_size, void* d_ws, size_t ws_size,
                              hipStream_t stream) {
  const float* q   = (const float*)d_in[0];
  const float* a   = (const float*)d_in[1];
  const float* nrm = (const float*)d_in[2];
  float* out = (float*)d_out;

  hipMemsetAsync(out, 0, (size_t)out_size * sizeof(float), stream);

  dim3 grid(B_ * (TPB / WPB));   // 4 * 64 = 256 blocks
  dim3 block(256);               // 8 waves
  size_t ldsBytes = (size_t)NAP * 4 * sizeof(float);  // 110,336 B (< 320 KB WGP LDS)
  hipLaunchKernelGGL(collision_count_wmma, grid, block, ldsBytes, stream,
                     q, a, nrm, out);
}